// GATLayer_3968549782111
// MI455X (gfx1250) — compile-verified
//
#include <hip/hip_runtime.h>
#include <hip/hip_bf16.h>

#define IN_F 128
#define OUT_F 32
#define NEG_SLOPE 0.2f

typedef float v2f __attribute__((ext_vector_type(2)));
typedef float v8f __attribute__((ext_vector_type(8)));

// ---------------------------------------------------------------------------
// Kernel 1: z = h @ W^T  via V_WMMA_F32_16X16X4_F32 (full fp32 precision).
// One wave32 handles a 16-row tile of h and all 32 output columns
// (two 16x16 f32 accumulators). K=128 consumed in steps of 4.
//
// ISA layouts (CDNA5 ISA):
//   A (16x4, MxK): lanes 0-15 -> K=kk,kk+1 in vgpr0,1 ; lanes 16-31 -> K=kk+2,kk+3.
//                  M = lane & 15 for all lanes.
//   B (4x16, KxN): vgpr0: K=kk (lanes 0-15) / K=kk+2 (lanes 16-31);
//                  vgpr1: K=kk+1 / K=kk+3.  N = lane & 15.  B[k][n] = W[n][k].
//   C/D (16x16):   vgpr r: M=r (lanes 0-15) / M=r+8 (lanes 16-31), N = lane & 15.
// ---------------------------------------------------------------------------
__global__ void __launch_bounds__(256) gat_gemm_wmma(
    const float* __restrict__ h, const float* __restrict__ W,
    float* __restrict__ z, int n_nodes)
{
    const int wave = threadIdx.x >> 5;
    const int lane = threadIdx.x & 31;
    const int tile = blockIdx.x * 8 + wave;      // 16-row tile index (wave-uniform)
    const int row0 = tile * 16;
    if (row0 >= n_nodes) return;                 // wave-uniform exit: EXEC all-1s for WMMA

    const int half = lane >> 4;                  // selects K sub-pair
    const int l16  = lane & 15;
    int arow = row0 + l16;
    if (arow >= n_nodes) arow = n_nodes - 1;     // clamp tail loads (stores guarded below)

    const float* __restrict__ hrow = h + (size_t)arow * IN_F;
    const float* __restrict__ w0   = W + (size_t)l16 * IN_F;        // out cols 0..15
    const float* __restrict__ w1   = W + (size_t)(l16 + 16) * IN_F; // out cols 16..31

    v8f acc0 = {};
    v8f acc1 = {};

#pragma unroll 4
    for (int kk = 0; kk < IN_F; kk += 4) {
        const int ka = kk + 2 * half;
        v2f A, B0, B1;
        A[0]  = hrow[ka];  A[1]  = hrow[ka + 1];
        B0[0] = w0[ka];    B0[1] = w0[ka + 1];
        B1[0] = w1[ka];    B1[1] = w1[ka + 1];
        acc0 = __builtin_amdgcn_wmma_f32_16x16x4_f32(false, A, false, B0,
                                                     (short)0, acc0, false, false);
        acc1 = __builtin_amdgcn_wmma_f32_16x16x4_f32(false, A, false, B1,
                                                     (short)0, acc1, false, false);
    }

#pragma unroll
    for (int r = 0; r < 8; ++r) {
        const int m = row0 + r + 8 * half;
        if (m < n_nodes) {
            z[(size_t)m * OUT_F + l16]      = acc0[r];
            z[(size_t)m * OUT_F + 16 + l16] = acc1[r];
        }
    }
}

// ---------------------------------------------------------------------------
// Kernel 2: per-node logits el = z.a_l, er = z.a_r; init emax=-inf, denom=0,
// out = 0 (harness poisons d_out/d_ws, so we must establish zeros every call).
// ---------------------------------------------------------------------------
__global__ void __launch_bounds__(256) gat_node_init(
    const float* __restrict__ z, const float* __restrict__ a,
    float* __restrict__ el, float* __restrict__ er,
    float* __restrict__ emax, float* __restrict__ denom,
    float* __restrict__ out, int n_nodes)
{
    const int n = blockIdx.x * blockDim.x + threadIdx.x;
    if (n >= n_nodes) return;
    const float* zr = z + (size_t)n * OUT_F;
    float sl = 0.f, sr = 0.f;
#pragma unroll
    for (int j = 0; j < OUT_F; ++j) {
        const float zv = zr[j];
        sl += zv * a[j];
        sr += zv * a[OUT_F + j];
    }
    el[n]    = sl;
    er[n]    = sr;
    emax[n]  = -__builtin_inff();
    denom[n] = 0.f;
    float4* o = (float4*)(out + (size_t)n * OUT_F);
#pragma unroll
    for (int j = 0; j < OUT_F / 4; ++j) o[j] = make_float4(0.f, 0.f, 0.f, 0.f);
}

// Float atomic max via sign-split int/uint atomics (valid with -inf init).
__device__ __forceinline__ void atomic_max_f32(float* addr, float v)
{
    if (v >= 0.f)
        atomicMax((int*)addr, __float_as_int(v));
    else
        atomicMin((unsigned int*)addr, __float_as_uint(v));
}

__device__ __forceinline__ float leaky(float v)
{
    return v > 0.f ? v : NEG_SLOPE * v;
}

// ---------------------------------------------------------------------------
// Kernel 3: segment max of edge logits over dst.
// ---------------------------------------------------------------------------
__global__ void __launch_bounds__(256) gat_edge_max(
    const int* __restrict__ src, const int* __restrict__ dst,
    const float* __restrict__ el, const float* __restrict__ er,
    float* __restrict__ emax, int n_edges)
{
    const int e = blockIdx.x * blockDim.x + threadIdx.x;
    if (e >= n_edges) return;
    const int s = src[e], d = dst[e];
    atomic_max_f32(&emax[d], leaky(el[s] + er[d]));
}

// ---------------------------------------------------------------------------
// Kernel 4: denom[d] = sum exp(e - emax[d]).
// ---------------------------------------------------------------------------
__global__ void __launch_bounds__(256) gat_edge_expsum(
    const int* __restrict__ src, const int* __restrict__ dst,
    const float* __restrict__ el, const float* __restrict__ er,
    const float* __restrict__ emax, float* __restrict__ denom, int n_edges)
{
    const int e = blockIdx.x * blockDim.x + threadIdx.x;
    if (e >= n_edges) return;
    const int s = src[e], d = dst[e];
    const float ex = __expf(leaky(el[s] + er[d]) - emax[d]);
    atomicAdd(&denom[d], ex);
}

// ---------------------------------------------------------------------------
// Kernel 5: out[d] += alpha * z[s].  8 threads per edge, float4 gather of z
// (L2-resident: z is 12.8 MB vs 192 MB L2), 4 fp32 atomic adds each.
// ---------------------------------------------------------------------------
__global__ void __launch_bounds__(256) gat_edge_aggregate(
    const int* __restrict__ src, const int* __restrict__ dst,
    const float* __restrict__ el, const float* __restrict__ er,
    const float* __restrict__ emax, const float* __restrict__ denom,
    const float* __restrict__ z, float* __restrict__ out, long n_edges)
{
    const long t = (long)blockIdx.x * blockDim.x + threadIdx.x;
    const long e = t >> 3;
    if (e >= n_edges) return;
    const int c = ((int)t & 7) * 4;
    const int s = src[e], d = dst[e];
    const float ex    = __expf(leaky(el[s] + er[d]) - emax[d]);
    const float alpha = ex / fmaxf(denom[d], 1e-16f);
    const float4 zc = *(const float4*)(z + (size_t)s * OUT_F + c);
    float* ob = out + (size_t)d * OUT_F + c;
    atomicAdd(ob + 0, alpha * zc.x);
    atomicAdd(ob + 1, alpha * zc.y);
    atomicAdd(ob + 2, alpha * zc.z);
    atomicAdd(ob + 3, alpha * zc.w);
}

// ---------------------------------------------------------------------------
extern "C" void kernel_launch(void* const* d_in, const int* in_sizes, int n_in,
                              void* d_out, int out_size, void* d_ws, size_t ws_size,
                              hipStream_t stream)
{
    const float* h  = (const float*)d_in[0];
    const float* W  = (const float*)d_in[1];
    const float* a  = (const float*)d_in[2];
    const int* src  = (const int*)d_in[3];
    const int* dst  = (const int*)d_in[4];
    float* out      = (float*)d_out;

    const int n_nodes = in_sizes[0] / IN_F;
    const int n_edges = in_sizes[3];

    // Workspace layout: z (N*32) | el (N) | er (N) | emax (N) | denom (N)
    float* z     = (float*)d_ws;
    float* el    = z + (size_t)n_nodes * OUT_F;
    float* er    = el + n_nodes;
    float* emax  = er + n_nodes;
    float* denom = emax + n_nodes;

    const int B = 256;
    const int ntiles = (n_nodes + 15) / 16;

    gat_gemm_wmma<<<dim3((ntiles + 7) / 8), dim3(B), 0, stream>>>(h, W, z, n_nodes);

    gat_node_init<<<dim3((n_nodes + B - 1) / B), dim3(B), 0, stream>>>(
        z, a, el, er, emax, denom, out, n_nodes);

    gat_edge_max<<<dim3((n_edges + B - 1) / B), dim3(B), 0, stream>>>(
        src, dst, el, er, emax, n_edges);

    gat_edge_expsum<<<dim3((n_edges + B - 1) / B), dim3(B), 0, stream>>>(
        src, dst, el, er, emax, denom, n_edges);

    const long t5 = (long)n_edges * 8;
    gat_edge_aggregate<<<dim3((unsigned)((t5 + B - 1) / B)), dim3(B), 0, stream>>>(
        src, dst, el, er, emax, denom, z, out, (long)n_edges);
}